// O_SVGD_68015102100170
// MI455X (gfx1250) — compile-verified
//
#include <hip/hip_runtime.h>
#include <math.h>

typedef __attribute__((ext_vector_type(2))) float v2f;
typedef __attribute__((ext_vector_type(8))) float v8f;

#define NPART 64
#define DDIM  32
#define NTHREADS 1024

__global__ __launch_bounds__(NTHREADS)
void osvgd_step(const float* __restrict__ particle,
                const float* __restrict__ mu,
                float* __restrict__ out)
{
    __shared__ float s_x[NPART * DDIM];     // particles
    __shared__ float s_u[NPART * DDIM];     // unit directions x/||x||
    __shared__ float s_K[NPART * NPART];    // dist2, then exp kernel (in place)
    __shared__ float s_P[NPART * NPART];    // sort scratch, then Gram x_i·x_j
    __shared__ float s_y[NPART * DDIM];     // y = K @ Ds (WMMA result)
    __shared__ float s_mu[DDIM];
    __shared__ float s_rs[NPART];           // 1/||x||
    __shared__ float s_rinv[NPART];         // 1/||x||^2
    __shared__ float s_ud[NPART];           // u_j · s_j
    __shared__ float s_uy[NPART];           // u_i · y_i
    __shared__ float s_denom;

    const int tid = threadIdx.x;

    // ---- phase 0: load inputs into LDS --------------------------------
    for (int e = tid; e < NPART * DDIM; e += NTHREADS) s_x[e] = particle[e];
    if (tid < DDIM) s_mu[tid] = mu[tid];
    __syncthreads();

    // ---- phase 1: per-particle norms ----------------------------------
    if (tid < NPART) {
        float r = 0.f;
        for (int c = 0; c < DDIM; ++c) { float v = s_x[tid * DDIM + c]; r += v * v; }
        s_rinv[tid] = 1.0f / r;
        s_rs[tid]   = rsqrtf(r);
    }
    __syncthreads();

    // ---- phase 2: unit dirs + pairwise dist2 (two copies) -------------
    for (int e = tid; e < NPART * DDIM; e += NTHREADS)
        s_u[e] = s_x[e] * s_rs[e >> 5];
    for (int e = tid; e < NPART * NPART; e += NTHREADS) {
        int i = e >> 6, j = e & 63;
        float d2 = 0.f;
        for (int c = 0; c < DDIM; ++c) {
            float df = s_x[i * DDIM + c] - s_x[j * DDIM + c];
            d2 += df * df;
        }
        s_K[e] = d2;
        s_P[e] = d2;
    }
    __syncthreads();

    // ---- phase 3: exact median via bitonic sort of 4096 floats --------
    for (unsigned k = 2; k <= 4096u; k <<= 1) {
        for (unsigned j = k >> 1; j > 0; j >>= 1) {
            for (unsigned e = tid; e < 4096u; e += NTHREADS) {
                unsigned x = e ^ j;
                if (x > e) {
                    float a = s_P[e], b = s_P[x];
                    bool up = ((e & k) == 0u);
                    if ((a > b) == up) { s_P[e] = b; s_P[x] = a; }
                }
            }
            __syncthreads();
        }
    }
    if (tid == 0) {
        float med = 0.5f * (s_P[2047] + s_P[2048]);   // even count: mean of middles
        s_denom = med / 4.1588830833596715f + 1e-6f;  // log(64)
    }
    __syncthreads();
    const float denom = s_denom;

    // ---- phase 4: K = exp(-d2/denom) in place; Gram P = x_i·x_j -------
    for (int e = tid; e < NPART * NPART; e += NTHREADS)
        s_K[e] = expf(-s_K[e] / denom);
    for (int e = tid; e < NPART * NPART; e += NTHREADS) {
        int i = e >> 6, j = e & 63;
        float p = 0.f;
        for (int c = 0; c < DDIM; ++c) p += s_x[i * DDIM + c] * s_x[j * DDIM + c];
        s_P[e] = p;
    }
    // u_j · (mu - x_j), for Ds fragments synthesized on the fly
    if (tid < NPART) {
        float a = 0.f;
        for (int c = 0; c < DDIM; ++c)
            a += s_u[tid * DDIM + c] * (s_mu[c] - s_x[tid * DDIM + c]);
        s_ud[tid] = a;
    }
    __syncthreads();

    // ---- phase 5: y = K @ Ds via v_wmma_f32_16x16x4_f32 ---------------
    // 64x64 @ 64x32 -> 8 tiles of 16x16; one wave per tile, 16 k-steps.
    if (tid < 256) {
        const int wave = tid >> 5;
        const int lane = tid & 31;
        const int mt   = wave >> 1;        // 0..3 row tile
        const int nt   = wave & 1;         // 0..1 col tile
        const int lo   = lane & 15;
        const int half = lane >> 4;        // 0/1
        v8f acc = {};
        for (int kk = 0; kk < 16; ++kk) {
            const int kb = kk * 4 + half * 2;
            v2f a, b;
            // A fragment: K tile rows, 16x4 f32 layout (lanes 0-15: k0/k1, 16-31: k2/k3)
            a.x = s_K[(mt * 16 + lo) * 64 + kb + 0];
            a.y = s_K[(mt * 16 + lo) * 64 + kb + 1];
            // B fragment: Ds[k][n] = (mu - x)[k][n] - u[k][n]*(u_k·s_k), built in-regs
            const int n = nt * 16 + lo;
            b.x = (s_mu[n] - s_x[(kb + 0) * DDIM + n]) - s_u[(kb + 0) * DDIM + n] * s_ud[kb + 0];
            b.y = (s_mu[n] - s_x[(kb + 1) * DDIM + n]) - s_u[(kb + 1) * DDIM + n] * s_ud[kb + 1];
            acc = __builtin_amdgcn_wmma_f32_16x16x4_f32(
                false, a, false, b, (short)0, acc, false, false);
        }
        for (int v = 0; v < 8; ++v) {
            const int m = mt * 16 + v + half * 8;
            const int n = nt * 16 + lo;
            s_y[m * DDIM + n] = acc[v];
        }
    }
    __syncthreads();

    // ---- phase 6: u_i · y_i -------------------------------------------
    if (tid < NPART) {
        float a = 0.f;
        for (int c = 0; c < DDIM; ++c) a += s_u[tid * DDIM + c] * s_y[tid * DDIM + c];
        s_uy[tid] = a;
    }
    __syncthreads();

    // ---- phase 7: analytic jacfwd-trace term + mle + step -------------
    // t_ij = k_ij [ -(2/denom)(u_j·x_i)(u_j - u_i(u_i·u_j))
    //              -(d-1)/r_j (x_j - u_i(u_i·x_j)) ]   (i != j)
    // t_ii = -(d-1)/r_i * x_i
    const float c2 = 2.0f / denom;
    for (int e = tid; e < NPART * DDIM; e += NTHREADS) {
        const int i = e >> 5, c = e & 31;
        const float xi  = s_x[e];
        const float ui  = s_u[e];
        const float rsi = s_rs[i];
        float acc = -31.0f * s_rinv[i] * xi;           // diagonal term
        for (int j = 0; j < NPART; ++j) {
            if (j == i) continue;
            const float kk  = s_K[i * 64 + j];
            const float p   = s_P[i * 64 + j];
            const float rsj = s_rs[j];
            const float g1  = p * rsj;                 // u_j · x_i
            const float g2  = g1 * rsi;                // u_i · u_j
            const float g1t = p * rsi;                 // u_i · x_j
            const float t = -c2 * g1 * (s_u[j * DDIM + c] - ui * g2)
                            - 31.0f * s_rinv[j] * (s_x[j * DDIM + c] - ui * g1t);
            acc += kk * t;
        }
        const float mle = s_y[e] - ui * s_uy[i];       // D_i applied to y_i
        const float g   = (mle + acc) * (1.0f / 64.0f);
        const float dx  = fminf(fmaxf(g * 0.1f, -1000.0f), 1000.0f);
        out[e] = xi + dx;
    }
}

extern "C" void kernel_launch(void* const* d_in, const int* in_sizes, int n_in,
                              void* d_out, int out_size, void* d_ws, size_t ws_size,
                              hipStream_t stream) {
    (void)in_sizes; (void)n_in; (void)out_size; (void)d_ws; (void)ws_size;
    const float* particle = (const float*)d_in[0];
    const float* mu       = (const float*)d_in[1];
    float*       out      = (float*)d_out;
    osvgd_step<<<1, NTHREADS, 0, stream>>>(particle, mu, out);
}